// BrainGNN_88785563943646
// MI455X (gfx1250) — compile-verified
//
#include <hip/hip_runtime.h>
#include <hip/hip_bf16.h>

// ---------------- problem constants (from reference) ----------------
#define BG      256        // graphs
#define NPG     400        // nodes per graph
#define KKEEP   200        // kept nodes per graph
#define EPG     8000
#define NN      (BG * NPG)     // 102400 nodes
#define EE      (BG * EPG)     // 2048000 edges
#define CINF    400
#define HF      64
#define H2F     128

typedef float v2f __attribute__((ext_vector_type(2)));
typedef float v4f __attribute__((ext_vector_type(4)));
typedef float v8f __attribute__((ext_vector_type(8)));

// ---------------------------------------------------------------
// Register-blocked f32 WMMA GEMM: C[M,NCOL] = A[M,K] @ B[K,NCOL] (+bias)
// One wave computes (16*TM) rows x NCOL cols via V_WMMA_F32_16X16X4_F32.
// K, NCOL, TM are compile-time so B/C addressing folds into immediate
// offsets; A fragments are reused across TN column tiles and B fragments
// across TM row tiles.
// wave32 layout (ISA 7.12.2): lanes 0-15 hold K=k0..k0+1, lanes 16-31
// hold K=k0+2..k0+3; D rows r (lanes<16) / r+8 (lanes>=16).
// ---------------------------------------------------------------
template<int K, int NCOL, int TM>
__global__ void wmma_gemm_f32_t(const float* __restrict__ A,
                                const float* __restrict__ Bm,
                                const float* __restrict__ bias,
                                float* __restrict__ C,
                                int M, int lda)
{
    constexpr int TN = NCOL / 16;
    const int wave = blockIdx.x * (blockDim.x >> 5) + (threadIdx.x >> 5);
    const int lane = threadIdx.x & 31;
    const int rowBase = wave * (16 * TM);
    if (rowBase >= M) return;

    const int half = lane >> 4;     // 0: lanes 0-15, 1: lanes 16-31
    const int l    = lane & 15;

    const float* arow[TM];
#pragma unroll
    for (int t = 0; t < TM; ++t)
        arow[t] = A + (size_t)(rowBase + t * 16 + l) * lda + half * 2;
    const float* bbase = Bm + (size_t)(half * 2) * NCOL + l;

    v8f acc[TM][TN];
#pragma unroll
    for (int t = 0; t < TM; ++t)
#pragma unroll
        for (int n = 0; n < TN; ++n)
            acc[t][n] = (v8f){0.f, 0.f, 0.f, 0.f, 0.f, 0.f, 0.f, 0.f};

    for (int k0 = 0; k0 < K; k0 += 4) {
        if ((k0 & 63) == 0) {
#pragma unroll
            for (int t = 0; t < TM; ++t)
                __builtin_prefetch(arow[t] + k0 + 64, 0, 1);
        }
        v2f a[TM];
#pragma unroll
        for (int t = 0; t < TM; ++t)
            a[t] = *(const v2f*)(arow[t] + k0);        // A[row][k0+2h .. +1]
        v2f b[TN];
#pragma unroll
        for (int n = 0; n < TN; ++n) {
            b[n].x = bbase[(size_t)k0 * NCOL + n * 16];        // B[k0+2h  ][col]
            b[n].y = bbase[(size_t)(k0 + 1) * NCOL + n * 16];  // B[k0+2h+1][col]
        }
#pragma unroll
        for (int t = 0; t < TM; ++t)
#pragma unroll
            for (int n = 0; n < TN; ++n)
                acc[t][n] = __builtin_amdgcn_wmma_f32_16x16x4_f32(
                    false, a[t], false, b[n], (short)0, acc[t][n], false, false);
    }

#pragma unroll
    for (int t = 0; t < TM; ++t)
#pragma unroll
        for (int n = 0; n < TN; ++n) {
            float bv = bias ? bias[n * 16 + l] : 0.f;
#pragma unroll
            for (int r = 0; r < 8; ++r) {
                int m = rowBase + t * 16 + half * 8 + r;
                C[(size_t)m * NCOL + n * 16 + l] = acc[t][n][r] + bv;
            }
        }
}

// ------------------- degree / normalization --------------------
__global__ void deg_init_kernel(float* __restrict__ deg,
                                const float* __restrict__ sl, int n)
{
    int i = blockIdx.x * blockDim.x + threadIdx.x;
    if (i < n) deg[i] = sl ? sl[i] : 1.0f;
}

__global__ void deg_edge_kernel(const int* __restrict__ dst,
                                const float* __restrict__ w,
                                float* __restrict__ deg, int nE)
{
    int e = blockIdx.x * blockDim.x + threadIdx.x;
    if (e < nE) {
        float we = w[e];
        if (we != 0.f) atomicAdd(&deg[dst[e]], we);
    }
}

__global__ void dis_kernel(const float* __restrict__ deg,
                           float* __restrict__ dis, int n)
{
    int i = blockIdx.x * blockDim.x + threadIdx.x;
    if (i < n) {
        float d = deg[i];
        dis[i] = (d > 0.f) ? rsqrtf(fmaxf(d, 1e-12f)) : 0.f;
    }
}

// ---- agg = dis^2 * sl * xW + bias  (self-loop term, full init) ----
__global__ void agg_init_kernel(const float* __restrict__ xw,
                                const float* __restrict__ dis,
                                const float* __restrict__ sl,    // nullable
                                const float* __restrict__ bias,
                                float* __restrict__ agg,
                                int fshift, long n)
{
    long idx = (long)blockIdx.x * blockDim.x + threadIdx.x;
    if (idx >= n) return;
    int node = (int)(idx >> fshift);
    int f    = (int)(idx & ((1 << fshift) - 1));
    float d  = dis[node];
    float s  = sl ? sl[node] : 1.0f;
    agg[idx] = d * d * s * xw[idx] + bias[f];
}

// ---- agg[dst] += dis[src]*w*dis[dst] * xW[src]  (edge scatter) ----
// thread = (edge, 16-feature chunk); vector b128 reads, f32 atomics (L2)
__global__ void edge_scatter_kernel(const int* __restrict__ src,
                                    const int* __restrict__ dst,
                                    const float* __restrict__ w,
                                    const float* __restrict__ dis,
                                    const float* __restrict__ xw,
                                    float* __restrict__ agg,
                                    int ldh, int cshift, long nWork)
{
    long t = (long)blockIdx.x * blockDim.x + threadIdx.x;
    if (t >= nWork) return;
    int e = (int)(t >> cshift);
    int c = (int)(t & ((1 << cshift) - 1));
    float we = w[e];
    if (we == 0.f) return;                 // dropped edges (layer 3)
    int s = src[e], d = dst[e];
    float norm = dis[s] * we * dis[d];
    if (norm == 0.f) return;
    const v4f* xr = (const v4f*)(xw + (size_t)s * ldh + c * 16);
    float* ar = agg + (size_t)d * ldh + c * 16;
#pragma unroll
    for (int q = 0; q < 4; ++q) {
        v4f v = xr[q];
        atomicAdd(&ar[q * 4 + 0], norm * v.x);
        atomicAdd(&ar[q * 4 + 1], norm * v.y);
        atomicAdd(&ar[q * 4 + 2], norm * v.z);
        atomicAdd(&ar[q * 4 + 3], norm * v.w);
    }
}

// ---- eval BatchNorm (mean0/var1) + ReLU, in place capable ----
__global__ void bn_relu_kernel(const float* __restrict__ in,
                               const float* __restrict__ g,
                               const float* __restrict__ bt,
                               float* __restrict__ out,
                               int fshift, long n)
{
    long idx = (long)blockIdx.x * blockDim.x + threadIdx.x;
    if (idx >= n) return;
    int f = (int)(idx & ((1 << fshift) - 1));
    float scale = rsqrtf(1.0f + 1e-5f);
    float v = in[idx] * (g[f] * scale) + bt[f];
    out[idx] = fmaxf(v, 0.f);
}

// ---- TopK score: tanh( h . p / ||p|| ) ----
__global__ void score_kernel(const float* __restrict__ h,
                             const float* __restrict__ p,
                             float* __restrict__ score, int n)
{
    int i = blockIdx.x * blockDim.x + threadIdx.x;
    if (i >= n) return;
    float nrm = 0.f;
#pragma unroll
    for (int k = 0; k < HF; ++k) nrm += p[k] * p[k];
    nrm = sqrtf(nrm);
    float dot = 0.f;
    const float* hr = h + (size_t)i * HF;
#pragma unroll
    for (int k = 0; k < HF; ++k) dot += hr[k] * p[k];
    score[i] = tanhf(dot / nrm);
}

// ---- per-graph top-K mask via stable rank count (one block / graph) ----
__global__ void topk_mask_kernel(const float* __restrict__ score,
                                 float* __restrict__ mask)
{
    __shared__ float s[NPG];
    int g = blockIdx.x;
    int i = threadIdx.x;
    if (i < NPG) s[i] = score[g * NPG + i];
    __syncthreads();
    if (i >= NPG) return;
    float si = s[i];
    int rank = 0;
    for (int j = 0; j < NPG; ++j) {
        float sj = s[j];
        rank += (sj > si) || (sj == si && j < i);
    }
    mask[g * NPG + i] = (rank < KKEEP) ? 1.0f : 0.0f;
}

// ---- gate node features: xg = h * (score*mask) ----
__global__ void gate_kernel(const float* __restrict__ h,
                            const float* __restrict__ score,
                            const float* __restrict__ mask,
                            float* __restrict__ xg, long n)
{
    long idx = (long)blockIdx.x * blockDim.x + threadIdx.x;
    if (idx >= n) return;
    int node = (int)(idx >> 6);            // H = 64
    xg[idx] = h[idx] * score[node] * mask[node];
}

// ---- gate edge weights: we = w * mask[src] * mask[dst] ----
__global__ void we_kernel(const int* __restrict__ src,
                          const int* __restrict__ dst,
                          const float* __restrict__ w,
                          const float* __restrict__ mask,
                          float* __restrict__ we, int nE)
{
    int e = blockIdx.x * blockDim.x + threadIdx.x;
    if (e < nE) we[e] = w[e] * mask[src[e]] * mask[dst[e]];
}

// ---- global mean/max pool over kept nodes (block = graph, thread = feat) ----
__global__ void pool_kernel(const float* __restrict__ h3,
                            const float* __restrict__ mask,
                            float* __restrict__ emb)
{
    int g = blockIdx.x;
    int f = threadIdx.x;                   // 0..127
    float sum = 0.f, mx = -INFINITY;
    for (int nl = 0; nl < NPG; ++nl) {
        int node = g * NPG + nl;
        float m = mask[node];
        float v = h3[(size_t)node * H2F + f];
        sum += m * v;
        if (m > 0.f) mx = fmaxf(mx, v);
    }
    emb[(size_t)g * (4 * HF) + f]       = sum / (float)KKEEP;
    emb[(size_t)g * (4 * HF) + H2F + f] = mx;
}

// ---- final tiny FC2: out[B,2] ----
__global__ void fc2_kernel(const float* __restrict__ z,
                           const float* __restrict__ w,
                           const float* __restrict__ b,
                           float* __restrict__ out)
{
    int t = blockIdx.x * blockDim.x + threadIdx.x;
    if (t >= BG * 2) return;
    int g = t >> 1, c = t & 1;
    float acc = b[c];
    const float* zr = z + g * 64;
#pragma unroll
    for (int k = 0; k < 64; ++k) acc += zr[k] * w[k * 2 + c];
    out[t] = acc;
}

// ---------------------------------------------------------------
extern "C" void kernel_launch(void* const* d_in, const int* in_sizes, int n_in,
                              void* d_out, int out_size, void* d_ws, size_t ws_size,
                              hipStream_t stream)
{
    const float* x      = (const float*)d_in[0];
    const int*   ei     = (const int*)  d_in[1];
    const int*   src    = ei;
    const int*   dst    = ei + EE;
    const float* ew     = (const float*)d_in[2];
    // d_in[3] = batch (implicit via block structure)
    const float* W1     = (const float*)d_in[4];
    const float* b1     = (const float*)d_in[5];
    const float* g1     = (const float*)d_in[6];
    const float* bt1    = (const float*)d_in[7];
    const float* W2     = (const float*)d_in[8];
    const float* b2     = (const float*)d_in[9];
    const float* g2     = (const float*)d_in[10];
    const float* bt2    = (const float*)d_in[11];
    const float* poolp  = (const float*)d_in[12];
    const float* W3     = (const float*)d_in[13];
    const float* b3     = (const float*)d_in[14];
    const float* g3     = (const float*)d_in[15];
    const float* bt3    = (const float*)d_in[16];
    const float* fc1w   = (const float*)d_in[17];
    const float* fc1b   = (const float*)d_in[18];
    const float* gfc    = (const float*)d_in[19];
    const float* bfc    = (const float*)d_in[20];
    const float* fc2w   = (const float*)d_in[21];
    const float* fc2b   = (const float*)d_in[22];
    float* out = (float*)d_out;

    // ---- workspace carve-up (all buffers fully written before read) ----
    float* xw   = (float*)d_ws;                       // N*2H (shared GEMM out)
    float* h1   = xw  + (size_t)NN * H2F;             // N*H   (later reused = xg)
    float* h2   = h1  + (size_t)NN * HF;              // N*H
    float* h3   = h2  + (size_t)NN * HF;              // N*2H
    float* deg  = h3  + (size_t)NN * H2F;             // N
    float* dis  = deg + NN;                           // N
    float* scr  = dis + NN;                           // N
    float* msk  = scr + NN;                           // N
    float* we   = msk + NN;                           // E
    float* emb  = we  + (size_t)EE;                   // B*4H
    float* z    = emb + (size_t)BG * 4 * HF;          // B*64
    float* xg   = h1;                                 // alias after layer 2

    const int T = 256;
    const long nH  = (long)NN * HF;
    const long nH2 = (long)NN * H2F;

    // ================= GCN layer 1 =================
    deg_init_kernel<<<(NN + T - 1) / T, T, 0, stream>>>(deg, nullptr, NN);
    deg_edge_kernel<<<(EE + T - 1) / T, T, 0, stream>>>(dst, ew, deg, EE);
    dis_kernel<<<(NN + T - 1) / T, T, 0, stream>>>(deg, dis, NN);

    {   // xw = x @ W1   (M=NN, NCOL=64, K=400), 32 rows/wave
        int waves = NN / 32;
        wmma_gemm_f32_t<CINF, HF, 2><<<(waves + 3) / 4, 128, 0, stream>>>(
            x, W1, nullptr, xw, NN, CINF);
    }
    agg_init_kernel<<<(nH + T - 1) / T, T, 0, stream>>>(xw, dis, nullptr, b1, h1, 6, nH);
    edge_scatter_kernel<<<(long)((EE * 4L + T - 1) / T), T, 0, stream>>>(
        src, dst, ew, dis, xw, h1, HF, 2, EE * 4L);
    bn_relu_kernel<<<(nH + T - 1) / T, T, 0, stream>>>(h1, g1, bt1, h1, 6, nH);

    // ================= GCN layer 2 ================= (same deg/dis)
    {   // xw = h1 @ W2  (K=64)
        int waves = NN / 32;
        wmma_gemm_f32_t<HF, HF, 2><<<(waves + 3) / 4, 128, 0, stream>>>(
            h1, W2, nullptr, xw, NN, HF);
    }
    agg_init_kernel<<<(nH + T - 1) / T, T, 0, stream>>>(xw, dis, nullptr, b2, h2, 6, nH);
    edge_scatter_kernel<<<(long)((EE * 4L + T - 1) / T), T, 0, stream>>>(
        src, dst, ew, dis, xw, h2, HF, 2, EE * 4L);
    bn_relu_kernel<<<(nH + T - 1) / T, T, 0, stream>>>(h2, g2, bt2, h2, 6, nH);

    // ================= TopK pooling =================
    score_kernel<<<(NN + T - 1) / T, T, 0, stream>>>(h2, poolp, scr, NN);
    topk_mask_kernel<<<BG, 512, 0, stream>>>(scr, msk);
    gate_kernel<<<(nH + T - 1) / T, T, 0, stream>>>(h2, scr, msk, xg, nH);
    we_kernel<<<(EE + T - 1) / T, T, 0, stream>>>(src, dst, ew, msk, we, EE);

    // ================= GCN layer 3 =================
    deg_init_kernel<<<(NN + T - 1) / T, T, 0, stream>>>(deg, msk, NN);
    deg_edge_kernel<<<(EE + T - 1) / T, T, 0, stream>>>(dst, we, deg, EE);
    dis_kernel<<<(NN + T - 1) / T, T, 0, stream>>>(deg, dis, NN);
    {   // xw = xg @ W3  (NCOL=128, K=64), 16 rows/wave (8 col tiles)
        int waves = NN / 16;
        wmma_gemm_f32_t<HF, H2F, 1><<<(waves + 3) / 4, 128, 0, stream>>>(
            xg, W3, nullptr, xw, NN, HF);
    }
    agg_init_kernel<<<(nH2 + T - 1) / T, T, 0, stream>>>(xw, dis, msk, b3, h3, 7, nH2);
    edge_scatter_kernel<<<(long)((EE * 8L + T - 1) / T), T, 0, stream>>>(
        src, dst, we, dis, xw, h3, H2F, 3, EE * 8L);
    bn_relu_kernel<<<(nH2 + T - 1) / T, T, 0, stream>>>(h3, g3, bt3, h3, 7, nH2);

    // ================= readout + MLP head =================
    pool_kernel<<<BG, H2F, 0, stream>>>(h3, msk, emb);
    {   // z = emb @ fc1_w + fc1_b   (M=256, NCOL=64, K=256)
        int waves = BG / 32;
        wmma_gemm_f32_t<4 * HF, HF, 2><<<(waves + 3) / 4, 128, 0, stream>>>(
            emb, fc1w, fc1b, z, BG, 4 * HF);
    }
    bn_relu_kernel<<<((long)BG * HF + T - 1) / T, T, 0, stream>>>(
        z, gfc, bfc, z, 6, (long)BG * HF);
    fc2_kernel<<<(BG * 2 + T - 1) / T, T, 0, stream>>>(z, fc2w, fc2b, out);
}